// Net_56856777064586
// MI455X (gfx1250) — compile-verified
//
#include <hip/hip_runtime.h>

#define HDIM 128

typedef float v2f __attribute__((ext_vector_type(2)));
typedef float v8f __attribute__((ext_vector_type(8)));

// ---------------- degree: deg[i] = #edges with src==i ----------------
__global__ void deg_kernel(const int* __restrict__ src, float* __restrict__ deg, int E) {
    int e = blockIdx.x * blockDim.x + threadIdx.x;
    if (e < E) atomicAdd(&deg[src[e]], 1.0f);
}

// ------------- scatter: Az[dst] += z[src], one wave per edge -------------
__global__ void scatter_kernel(const int* __restrict__ src, const int* __restrict__ dst,
                               const float* __restrict__ z, float* __restrict__ Az, int E) {
    long t = (long)blockIdx.x * blockDim.x + threadIdx.x;
    int e    = (int)(t >> 5);
    int lane = (int)(t & 31);
    if (e >= E) return;
    int s = src[e], d = dst[e];
    float4 v = *(const float4*)&z[(long)s * HDIM + lane * 4];
    float* base = &Az[(long)d * HDIM + lane * 4];
    atomicAdd(base + 0, v.x);
    atomicAdd(base + 1, v.y);
    atomicAdd(base + 2, v.z);
    atomicAdd(base + 3, v.w);
}

// ------------- Tx1 = scale*(deg*x - Az) - x -------------
__global__ void cheb1_kernel(const float* __restrict__ x, const float* __restrict__ Az,
                             const float* __restrict__ deg, const float* __restrict__ lam,
                             float* __restrict__ t1, int N) {
    long i = (long)blockIdx.x * blockDim.x + threadIdx.x;   // index in float4 units
    long total = (long)N * (HDIM / 4);
    if (i >= total) return;
    int node = (int)(i >> 5);
    float scale = 2.0f / lam[0];
    float dg = deg[node];
    float4 xv = ((const float4*)x)[i];
    float4 av = ((const float4*)Az)[i];
    float4 r;
    r.x = scale * (dg * xv.x - av.x) - xv.x;
    r.y = scale * (dg * xv.y - av.y) - xv.y;
    r.z = scale * (dg * xv.z - av.z) - xv.z;
    r.w = scale * (dg * xv.w - av.w) - xv.w;
    ((float4*)t1)[i] = r;
}

// ------------- Tx2 = 2*(scale*(deg*t1 - Az1) - t1) - x0 -------------
__global__ void cheb2_kernel(const float* __restrict__ x0, const float* __restrict__ t1,
                             const float* __restrict__ Az, const float* __restrict__ deg,
                             const float* __restrict__ lam, float* __restrict__ t2, int N) {
    long i = (long)blockIdx.x * blockDim.x + threadIdx.x;
    long total = (long)N * (HDIM / 4);
    if (i >= total) return;
    int node = (int)(i >> 5);
    float scale = 2.0f / lam[0];
    float dg = deg[node];
    float4 xv = ((const float4*)x0)[i];
    float4 tv = ((const float4*)t1)[i];
    float4 av = ((const float4*)Az)[i];
    float4 r;
    r.x = 2.0f * (scale * (dg * tv.x - av.x) - tv.x) - xv.x;
    r.y = 2.0f * (scale * (dg * tv.y - av.y) - tv.y) - xv.y;
    r.z = 2.0f * (scale * (dg * tv.z - av.z) - tv.z) - xv.z;
    r.w = 2.0f * (scale * (dg * tv.w - av.w) - tv.w) - xv.w;
    ((float4*)t2)[i] = r;
}

// ------------- fused triple GEMM + bias + ReLU -------------
// Xout = relu(T0*W[0] + T1*W[1] + T2*W[2] + bias), W row-major [k][col].
// 256 threads = 8 waves; wave w owns rows [blk*128 + 16w, +16), all 128 cols.
// V_WMMA_F32_16X16X4_F32 layouts:
//   A 16x4: lanes 0-15 hold K=0/1 in v[0]/v[1], lanes 16-31 hold K=2/3.
//   B 4x16: symmetric (VGPR0: K=0 lanes 0-15 / K=2 lanes 16-31; VGPR1: K=1/K=3).
//   C/D 16x16: VGPR v holds rows v (lanes 0-15) and v+8 (lanes 16-31).
// W is staged into LDS in a fragment-ready layout so each B fragment is one
// contiguous 8-byte ds_load_b64: element (kk,col) -> lds[(kk>>1)*256 + col*2 + (kk&1)].
__global__ __launch_bounds__(256)
void gemm3_kernel(const float* T0, const float* T1, const float* T2,
                  const float* __restrict__ W, const float* __restrict__ bias,
                  float* Xout, int N) {
    __shared__ float ldsW[HDIM * HDIM];   // 64 KB, swizzled layout

    const int tid  = threadIdx.x;
    const int wave = tid >> 5;
    const int lane = tid & 31;
    const int l16  = lane & 15;
    const int hi   = lane >> 4;           // 0 or 1
    const int koff = hi << 1;             // 0 or 2
    const int rowBase = blockIdx.x * 128 + wave * 16;

    v8f acc[8] = {};

    const float* Ts[3] = {T0, T1, T2};

    for (int c = 0; c < 3; ++c) {
        // stage W_c into LDS with fragment-ready swizzle
        const float4* Wc = (const float4*)(W + (size_t)c * HDIM * HDIM);
        for (int i = tid; i < HDIM * HDIM / 4; i += 256) {
            float4 v = Wc[i];
            int kk   = i >> 5;            // row (32 float4 per 128-wide row)
            int col0 = (i & 31) << 2;
            int base = (kk >> 1) * (HDIM * 2) + (kk & 1);
            ldsW[base + (col0 + 0) * 2] = v.x;
            ldsW[base + (col0 + 1) * 2] = v.y;
            ldsW[base + (col0 + 2) * 2] = v.z;
            ldsW[base + (col0 + 3) * 2] = v.w;
        }
        __syncthreads();

        int r = rowBase + l16;
        if (r > N - 1) r = N - 1;         // clamp: garbage rows never stored
        const float* Trow = Ts[c] + (long)r * HDIM + koff;

        #pragma unroll 4
        for (int k = 0; k < HDIM; k += 4) {
            v2f a = *(const v2f*)&Trow[k];   // A frag: K = k+koff, k+koff+1
            // B frag base for this wave-half: kpair = (k+koff)/2
            const float* bbase = &ldsW[((k + koff) >> 1) * (HDIM * 2) + l16 * 2];
            #pragma unroll
            for (int t = 0; t < 8; ++t) {
                v2f b = *(const v2f*)&bbase[t * 32];   // contiguous (kk,kk+1) pair
                acc[t] = __builtin_amdgcn_wmma_f32_16x16x4_f32(
                    false, a, false, b, (short)0, acc[t], false, false);
            }
        }
        __syncthreads();
    }

    // epilogue: bias + relu + store (safe even if Xout aliases T1: per-wave rows)
    #pragma unroll
    for (int t = 0; t < 8; ++t) {
        int col = t * 16 + l16;
        float bv = bias[col];
        #pragma unroll
        for (int v = 0; v < 8; ++v) {
            int row = rowBase + v + (hi << 3);
            if (row < N) {
                float val = acc[t][v] + bv;
                Xout[(long)row * HDIM + col] = val > 0.0f ? val : 0.0f;
            }
        }
    }
}

// ------------- final matvec: out[n] = dot(X[n,:], Wf) + bf -------------
__global__ void matvec_kernel(const float* __restrict__ X, const float* __restrict__ Wf,
                              const float* __restrict__ bf, float* __restrict__ out, int N) {
    long t = (long)blockIdx.x * blockDim.x + threadIdx.x;
    int n    = (int)(t >> 5);
    int lane = (int)(t & 31);
    if (n >= N) return;
    float4 xv = *(const float4*)&X[(long)n * HDIM + lane * 4];
    float4 wv = *(const float4*)&Wf[lane * 4];
    float s = xv.x * wv.x + xv.y * wv.y + xv.z * wv.z + xv.w * wv.w;
    #pragma unroll
    for (int off = 16; off > 0; off >>= 1) s += __shfl_xor(s, off, 32);
    if (lane == 0) out[n] = s + bf[0];
}

extern "C" void kernel_launch(void* const* d_in, const int* in_sizes, int n_in,
                              void* d_out, int out_size, void* d_ws, size_t ws_size,
                              hipStream_t stream) {
    const float* x   = (const float*)d_in[0];
    const int*   ei  = (const int*)d_in[1];
    const float* lam = (const float*)d_in[2];
    const float* Ws  = (const float*)d_in[3];
    const float* bs  = (const float*)d_in[4];
    const float* Wf  = (const float*)d_in[5];
    const float* bf  = (const float*)d_in[6];
    float* out = (float*)d_out;

    const int N = in_sizes[0] / HDIM;       // 100000
    const int E = in_sizes[1] / 2;          // 1600000
    const int* src = ei;
    const int* dst = ei + E;

    const size_t NH = (size_t)N * HDIM;
    float* wsf = (float*)d_ws;
    float* deg = wsf;
    float* Az  = wsf + (((size_t)N + 127) & ~(size_t)127);
    float* P   = Az + NH;
    float* Q   = P + NH;
    float* R   = Q + NH;

    // degree
    hipMemsetAsync(deg, 0, (size_t)N * sizeof(float), stream);
    deg_kernel<<<(E + 255) / 256, 256, 0, stream>>>(src, deg, E);

    const int scatterBlocks = (int)(((long)E * 32 + 255) / 256);
    const int elemBlocks    = (int)((NH / 4 + 255) / 256);
    const int gemmBlocks    = (N + 127) / 128;

    // per-block buffer rotation: {t1, t2, xout}; xout aliases t1 (safe, see kernel)
    float* rot[3][3] = { {P, Q, P}, {Q, R, Q}, {R, P, R} };

    const float* T0 = x;
    for (int b = 0; b < 3; ++b) {
        float* t1 = rot[b][0];
        float* t2 = rot[b][1];
        float* xo = rot[b][2];

        hipMemsetAsync(Az, 0, NH * sizeof(float), stream);
        scatter_kernel<<<scatterBlocks, 256, 0, stream>>>(src, dst, T0, Az, E);
        cheb1_kernel<<<elemBlocks, 256, 0, stream>>>(T0, Az, deg, lam, t1, N);

        hipMemsetAsync(Az, 0, NH * sizeof(float), stream);
        scatter_kernel<<<scatterBlocks, 256, 0, stream>>>(src, dst, t1, Az, E);
        cheb2_kernel<<<elemBlocks, 256, 0, stream>>>(T0, t1, Az, deg, lam, t2, N);

        gemm3_kernel<<<gemmBlocks, 256, 0, stream>>>(
            T0, t1, t2, Ws + (size_t)b * 3 * HDIM * HDIM, bs + (size_t)b * HDIM, xo, N);
        T0 = xo;
    }

    matvec_kernel<<<(int)(((long)N * 32 + 255) / 256), 256, 0, stream>>>(T0, Wf, bf, out, N);
}